// encoder_80607946211781
// MI455X (gfx1250) — compile-verified
//
#include <hip/hip_runtime.h>

// ---------------------------------------------------------------------------
// DA-RNN input-attention encoder for MI455X (gfx1250, wave32).
//   B=128, Tm1=128, D=128, H=64, fp32.
//   K0 pack_weights : transpose GEMV weights (coalesced register fills).
//   K1 uex_wmma     : Ue_x = x_perm @ U_e_w^T + U_e_b via V_WMMA_F32_16X16X4_F32.
//   K2 scan         : 128 sequential steps, one WG per batch row.
//                     Persistent-RNN: each thread keeps its GEMV weight
//                     columns in VGPRs (256 regs) -> zero per-step weight
//                     memory traffic; only the 64KB Ue_x block streams from
//                     WGP$/L2 each step.
// ---------------------------------------------------------------------------

typedef __attribute__((ext_vector_type(2))) float v2f;
typedef __attribute__((ext_vector_type(8))) float v8f;

#define NB   128   // batch
#define NT   128   // Tm1
#define ND   128   // drivers
#define NH   64    // hidden
#define NG   256   // 4*H

__device__ __forceinline__ float fast_exp(float x) {
  return __builtin_amdgcn_exp2f(x * 1.4426950408889634f);
}
__device__ __forceinline__ float fast_tanh(float x) {
  // tanh(x) = 1 - 2/(e^{2x}+1); exp2 overflow->inf->rcp->0 gives exact +/-1 tails
  float e = __builtin_amdgcn_exp2f(x * 2.8853900817779268f);
  return 1.0f - 2.0f * __builtin_amdgcn_rcpf(e + 1.0f);
}
__device__ __forceinline__ float fast_sigmoid(float x) {
  return __builtin_amdgcn_rcpf(1.0f + __builtin_amdgcn_exp2f(-1.4426950408889634f * x));
}

// ---------------------------------------------------------------------------
// K0: weight repack.  WehsT[k][s] = W_ehs_w[s][k]   (128x128)
//                     WihT [k][g] = W_ih[g][k]      (128x256)
//                     WhhT [j][g] = W_hh[g][j]      (64x256)
//                     biasIF[g]   = b_ih[g]+b_hh[g] (256)
// ---------------------------------------------------------------------------
__global__ void pack_weights(const float* __restrict__ Wehs,
                             const float* __restrict__ Wih,
                             const float* __restrict__ Whh,
                             const float* __restrict__ bih,
                             const float* __restrict__ bhh,
                             float* __restrict__ WehsT, float* __restrict__ WihT,
                             float* __restrict__ WhhT, float* __restrict__ biasIF) {
  int idx = blockIdx.x * blockDim.x + threadIdx.x;
  if (idx < NT * 128) {
    int s = idx >> 7, k = idx & 127;
    WehsT[k * NT + s] = Wehs[s * 128 + k];
  }
  int i2 = idx - NT * 128;
  if (i2 >= 0 && i2 < NG * ND) {
    int g = i2 >> 7, k = i2 & 127;
    WihT[k * NG + g] = Wih[g * ND + k];
  }
  int i3 = idx - NT * 128 - NG * ND;
  if (i3 >= 0 && i3 < NG * NH) {
    int g = i3 >> 6, j = i3 & 63;
    WhhT[j * NG + g] = Whh[g * NH + j];
  }
  int i4 = idx - NT * 128 - NG * ND - NG * NH;
  if (i4 >= 0 && i4 < NG) biasIF[i4] = bih[i4] + bhh[i4];
}

// ---------------------------------------------------------------------------
// K1: Ue_x[r=b*128+d][s] = sum_t A[r][t] * U_e_w[s][t] + U_e_b[s]
//     A[r][t] = input[b][t][d]  (transposed view, b=r>>7, d=r&127)
//     V_WMMA_F32_16X16X4_F32, one wave per 16x16 tile, 8 waves/block over N.
// ---------------------------------------------------------------------------
__global__ __launch_bounds__(256) void uex_wmma(const float* __restrict__ in,
                                                const float* __restrict__ Uw,
                                                const float* __restrict__ Ub,
                                                float* __restrict__ uex) {
  const int lane  = threadIdx.x;        // 0..31
  const int tileN = threadIdx.y;        // 0..7
  const int tileM = blockIdx.x;         // 0..1023
  const int half  = lane >> 4;          // 0 / 1
  const int lrow  = lane & 15;
  const int klo   = half << 1;          // K offset 0 or 2

  const int r  = tileM * 16 + lrow;     // A row (16-row tiles never cross b)
  const int bb = r >> 7;
  const int dd = r & 127;
  const float* __restrict__ Abase = in + bb * (NT * ND) + dd;  // + t*ND
  const int s = tileN * 16 + lrow;      // B "column" = output col
  const float* __restrict__ Bbase = Uw + s * NT;

  v8f acc = {};
#pragma unroll 4
  for (int k0 = 0; k0 < NT; k0 += 4) {
    const int k = k0 + klo;
    if (k0 + 8 < NT) __builtin_prefetch(Abase + (k + 8) * ND, 0, 1);
    v2f a;
    a.x = Abase[k * ND];
    a.y = Abase[(k + 1) * ND];
    v2f bf;
    bf.x = Bbase[k];
    bf.y = Bbase[k + 1];
    acc = __builtin_amdgcn_wmma_f32_16x16x4_f32(false, a, false, bf,
                                                (short)0, acc, false, false);
  }

  const int n     = tileN * 16 + lrow;
  const float bias = Ub[n];
  const int mbase = tileM * 16 + half * 8;
#pragma unroll
  for (int i = 0; i < 8; ++i)
    uex[(mbase + i) * NT + n] = acc[i] + bias;
}

// ---------------------------------------------------------------------------
// K2: 128-step recurrence, one WG (256 threads = 8 wave32) per batch row.
//     GEMV weights live in VGPRs (persistent-RNN); LDS only holds the ~5KB
//     of per-step shared state; Ue_x (64KB/b) streams from WGP$/L2.
// ---------------------------------------------------------------------------
__global__ __launch_bounds__(256, 1) void scan_kernel(
    const float* __restrict__ in, const float* __restrict__ WehsT,
    const float* __restrict__ Wehsb, const float* __restrict__ vew,
    const float* __restrict__ veb, const float* __restrict__ WihT,
    const float* __restrict__ WhhT, const float* __restrict__ biasIF,
    const float* __restrict__ uex, float* __restrict__ out_w,
    float* __restrict__ out_h) {
  __shared__ __align__(16) float hcs[128];   // h[0..63] | c[64..127]
  __shared__ float we_s[NT];
  __shared__ float part[256];
  __shared__ float v_s[ND];
  __shared__ float a_s[ND];
  __shared__ __align__(16) float xh_s[ND];
  __shared__ float gate_s[NG];

  const int b    = blockIdx.x;
  const int tid  = threadIdx.x;
  const int lane = tid & 31;
  const int wave = tid >> 5;

  // ---- fill persistent register-resident weight columns (once) ----
  const int s128 = tid & 127;          // attention output index for phase 1
  const int khalf = (tid >> 7) * 64;   // this thread's K half for phase 1
  float wehs_r[64];                    // WehsT[khalf+k][s128]
#pragma unroll
  for (int k = 0; k < 64; ++k)
    wehs_r[k] = WehsT[(khalf + k) * NT + s128];

  float wih_r[128];                    // WihT[k][tid]
#pragma unroll
  for (int k = 0; k < 128; ++k)
    wih_r[k] = WihT[k * NG + tid];

  float whh_r[64];                     // WhhT[j][tid]
#pragma unroll
  for (int j = 0; j < 64; ++j)
    whh_r[j] = WhhT[j * NG + tid];

  const float bias_g = biasIF[tid];
  const float wbv    = Wehsb[s128];    // W_ehs bias for this thread's s
  const float vebv   = veb[0];
  // per-lane attention-score weights (lane covers t' = lane+32q)
  const float vw0 = vew[lane], vw1 = vew[lane + 32],
              vw2 = vew[lane + 64], vw3 = vew[lane + 96];

  if (tid < 128) hcs[tid] = 0.0f;
  const float* __restrict__ uex_b = uex + b * (ND * NT);
  const float* __restrict__ in_b  = in + b * (NT * ND);
  __syncthreads();

  for (int t = 0; t < NT; ++t) {
    // ---- phase 1: we[s] = sum_k hc[k]*Wehs[s][k] + b  (2 threads per s) ----
    {
      float acc = 0.0f;
#pragma unroll
      for (int k = 0; k < 64; k += 4) {
        const float4 hc4 = *reinterpret_cast<const float4*>(&hcs[khalf + k]);
        acc = fmaf(hc4.x, wehs_r[k],     acc);
        acc = fmaf(hc4.y, wehs_r[k + 1], acc);
        acc = fmaf(hc4.z, wehs_r[k + 2], acc);
        acc = fmaf(hc4.w, wehs_r[k + 3], acc);
      }
      part[tid] = acc;
    }
    __syncthreads();
    if (tid < 128) we_s[tid] = part[tid] + part[tid + 128] + wbv;
    __syncthreads();

    // ---- phase 2: v[d] = sum_t' tanh(we[t'] + Uex[d][t'])*v_e_w[t'] + v_e_b
    //      wave handles 16 d's; lane <-> t' (coalesced Uex); wave32 butterfly.
    {
      const float we0 = we_s[lane], we1 = we_s[lane + 32],
                  we2 = we_s[lane + 64], we3 = we_s[lane + 96];
      for (int i = 0; i < 16; ++i) {
        const int d = wave * 16 + i;
        const float* __restrict__ row = uex_b + d * NT;
        float acc = fast_tanh(we0 + row[lane]) * vw0;
        acc = fmaf(fast_tanh(we1 + row[lane + 32]), vw1, acc);
        acc = fmaf(fast_tanh(we2 + row[lane + 64]), vw2, acc);
        acc = fmaf(fast_tanh(we3 + row[lane + 96]), vw3, acc);
#pragma unroll
        for (int off = 16; off >= 1; off >>= 1)
          acc += __shfl_xor(acc, off, 32);
        if (lane == 0) v_s[d] = acc + vebv;
      }
    }
    __syncthreads();

    // ---- phase 3: softmax over d, x_hat = a * x_t ----
    if (tid < 128) part[tid] = v_s[tid];
    __syncthreads();
    for (int off = 64; off >= 1; off >>= 1) {
      if (tid < off) part[tid] = fmaxf(part[tid], part[tid + off]);
      __syncthreads();
    }
    const float vmax = part[0];
    __syncthreads();
    if (tid < 128) { float e = fast_exp(v_s[tid] - vmax); a_s[tid] = e; part[tid] = e; }
    __syncthreads();
    for (int off = 64; off >= 1; off >>= 1) {
      if (tid < off) part[tid] += part[tid + off];
      __syncthreads();
    }
    const float rsum = 1.0f / part[0];
    __syncthreads();
    if (tid < 128) {
      const float xh = a_s[tid] * rsum * in_b[t * ND + tid];
      xh_s[tid] = xh;
      out_w[b * (NT * ND) + t * ND + tid] = xh;  // input_weighted
    }
    __syncthreads();

    // ---- phase 4: gates[g] = bias + x_hat@W_ih^T + h@W_hh^T; LSTM update ----
    {
      float acc = bias_g;
#pragma unroll
      for (int k = 0; k < ND; k += 4) {
        const float4 x4 = *reinterpret_cast<const float4*>(&xh_s[k]);
        acc = fmaf(x4.x, wih_r[k],     acc);
        acc = fmaf(x4.y, wih_r[k + 1], acc);
        acc = fmaf(x4.z, wih_r[k + 2], acc);
        acc = fmaf(x4.w, wih_r[k + 3], acc);
      }
#pragma unroll
      for (int j = 0; j < NH; j += 4) {
        const float4 h4 = *reinterpret_cast<const float4*>(&hcs[j]);
        acc = fmaf(h4.x, whh_r[j],     acc);
        acc = fmaf(h4.y, whh_r[j + 1], acc);
        acc = fmaf(h4.z, whh_r[j + 2], acc);
        acc = fmaf(h4.w, whh_r[j + 3], acc);
      }
      gate_s[tid] = acc;
    }
    __syncthreads();
    if (tid < NH) {
      const float ig = fast_sigmoid(gate_s[tid]);
      const float fg = fast_sigmoid(gate_s[NH + tid]);
      const float gg = fast_tanh(gate_s[2 * NH + tid]);
      const float og = fast_sigmoid(gate_s[3 * NH + tid]);
      const float cn = fg * hcs[64 + tid] + ig * gg;
      const float hn = og * fast_tanh(cn);
      out_h[(b * NT + t) * NH + tid] = hn;       // input_encoded
      hcs[tid] = hn;
      hcs[64 + tid] = cn;
    }
    __syncthreads();
  }
}

// ---------------------------------------------------------------------------
extern "C" void kernel_launch(void* const* d_in, const int* in_sizes, int n_in,
                              void* d_out, int out_size, void* d_ws, size_t ws_size,
                              hipStream_t stream) {
  const float* input = (const float*)d_in[0];   // [B, Tm1, D]
  const float* Wehs  = (const float*)d_in[1];   // [Tm1, 2H]
  const float* Wehsb = (const float*)d_in[2];   // [Tm1]
  const float* Uew   = (const float*)d_in[3];   // [Tm1, Tm1]
  const float* Ueb   = (const float*)d_in[4];   // [Tm1]
  const float* vew   = (const float*)d_in[5];   // [1, Tm1]
  const float* veb   = (const float*)d_in[6];   // [1]
  const float* Wih   = (const float*)d_in[7];   // [4H, D]
  const float* Whh   = (const float*)d_in[8];   // [4H, H]
  const float* bih   = (const float*)d_in[9];   // [4H]
  const float* bhh   = (const float*)d_in[10];  // [4H]

  float* ws     = (float*)d_ws;
  float* WehsT  = ws;                                   // 16384
  float* WihT   = ws + 16384;                           // 32768
  float* WhhT   = ws + 16384 + 32768;                   // 16384
  float* biasIF = ws + 16384 + 32768 + 16384;           // 256
  float* uexbuf = ws + 16384 + 32768 + 16384 + 256;     // 2,097,152 (8 MB)

  float* out_w = (float*)d_out;                 // [B, Tm1, D]
  float* out_h = out_w + NB * NT * ND;          // [B, Tm1, H]

  pack_weights<<<dim3(257), dim3(256), 0, stream>>>(Wehs, Wih, Whh, bih, bhh,
                                                    WehsT, WihT, WhhT, biasIF);
  uex_wmma<<<dim3(1024), dim3(32, 8), 0, stream>>>(input, Uew, Ueb, uexbuf);
  scan_kernel<<<dim3(NB), dim3(256), 0, stream>>>(input, WehsT, Wehsb, vew, veb,
                                                  WihT, WhhT, biasIF, uexbuf,
                                                  out_w, out_h);
}